// CustomMinkUNet14_74225624809958
// MI455X (gfx1250) — compile-verified
//
#include <hip/hip_runtime.h>

typedef float v2f __attribute__((ext_vector_type(2)));
typedef float v8f __attribute__((ext_vector_type(8)));

// ---------------------------------------------------------------------------
// Sparse gather-GEMM using V_WMMA_F32_16X16X4_F32 (f32 matches reference).
//   y[n, co] = sum_k mask[n,k] * sum_ci x[nbr[n,k], ci] * W[k, ci, co]
// nbr == nullptr -> identity gather (dense 1x1 matmul), mask = 1.
//
// One wave owns one 16-row output tile and COT=ceil(Co/16) accumulators, so
// each gathered A fragment (global_load_b64, rows are Ci%4==0 aligned) is
// reused across all output-channel tiles. Each 128-thread block (4 waves, 4
// row tiles) cooperatively stages W[k] (<= 24KB) into LDS in a pair-
// interleaved layout so B fragments are single aligned ds_load_b64 reads.
//
// A-fragment layout per CDNA5 ISA 7.12.2 (16x4 f32):
//   lane m (m<16):  a[0]=K=cb+0, a[1]=K=cb+1   (row M=m)
//   lane m+16:      a[0]=K=cb+2, a[1]=K=cb+3   (row M=m)
// B mirrored (column N=lane&15); C/D: acc[r] -> (row=rt*16+r+8*half, col).
// ---------------------------------------------------------------------------
template <int COT, bool CI4>
__global__ __launch_bounds__(128) void sconv16(
    const float* __restrict__ x, const int* __restrict__ nbr,
    const float* __restrict__ msk, const float* __restrict__ W,
    float* __restrict__ y, int Nout, int K, int Ci, int Co) {
  __shared__ float wlds[6400];  // >= max Ci*Co (96*64=6144) + slack
  const int tid = threadIdx.x;
  const int lane = tid & 31;
  const int wv = tid >> 5;
  const int rowTiles = (Nout + 15) >> 4;
  int rt = blockIdx.x * 4 + wv;
  const bool doStore = rt < rowTiles;
  if (!doStore) rt = rowTiles - 1;       // clamp: keep all waves in barriers
  const int m = lane & 15;
  const int half = lane >> 4;            // 0 or 1
  const int row = rt * 16 + m;
  const bool rvalid = row < Nout;
  const int rclamp = rvalid ? row : (Nout - 1);
  const bool ident = (nbr == nullptr);
  const int CiCo = Ci * Co;
  const int Co2 = Co << 1;

  v8f acc[COT];
#pragma unroll
  for (int t = 0; t < COT; ++t) acc[t] = (v8f){};

  for (int k = 0; k < K; ++k) {
    const float* __restrict__ Wk = W + (long)k * CiCo;
    if (CI4) {
      __syncthreads();
      for (int i = tid; i < CiCo; i += 128) {
        const int ci = i / Co, co = i - ci * Co;
        wlds[(ci >> 1) * Co2 + (co << 1) + (ci & 1)] = Wk[i];
      }
      __syncthreads();
      if (k + 1 < K) __builtin_prefetch(W + (long)(k + 1) * CiCo, 0, 3);
    }
    int idx;
    float mv;
    if (ident) {
      idx = rclamp;
      mv = rvalid ? 1.0f : 0.0f;
    } else {
      idx = nbr[rclamp * K + k];
      mv = rvalid ? msk[rclamp * K + k] : 0.0f;
    }
    const float* __restrict__ xr = x + (long)idx * Ci;

    for (int cb = 0; cb < Ci; cb += 4) {
      const int ka0 = cb + 2 * half;
      v2f a;
      if (CI4) {
        a = *(const v2f*)(xr + ka0);     // 8B-aligned contiguous gather
        a[0] *= mv;
        a[1] *= mv;
      } else {
        a[0] = (ka0 < Ci) ? xr[ka0] * mv : 0.0f;
        a[1] = (ka0 + 1 < Ci) ? xr[ka0 + 1] * mv : 0.0f;
      }
#pragma unroll
      for (int t = 0; t < COT; ++t) {
        const int cn = t * 16 + m;
        v2f b;
        if (CI4) {
          // pair-interleaved LDS: (ka0,ka0+1) adjacent -> one ds_load_b64.
          // cn >= Co reads stale data; those columns are never stored and
          // WMMA columns are independent, so it is harmless.
          b = *(const v2f*)&wlds[(ka0 >> 1) * Co2 + (cn << 1)];
        } else {
          b[0] = (cn < Co && ka0 < Ci) ? Wk[ka0 * Co + cn] : 0.0f;
          b[1] = (cn < Co && ka0 + 1 < Ci) ? Wk[(ka0 + 1) * Co + cn] : 0.0f;
        }
        acc[t] = __builtin_amdgcn_wmma_f32_16x16x4_f32(
            false, a, false, b, (short)0, acc[t], false, false);
      }
    }
  }

#pragma unroll
  for (int t = 0; t < COT; ++t) {
    const int cn = t * 16 + m;
    if (doStore && cn < Co) {
#pragma unroll
      for (int r = 0; r < 8; ++r) {
        const int orow = rt * 16 + r + 8 * half;
        if (orow < Nout) y[(long)orow * Co + cn] = acc[t][r];
      }
    }
  }
}

// ---------------------------------------------------------------------------
// BatchNorm (training) stats: one block per channel -> stats[c]=mean,
// stats[C+c]=rsqrt(var+eps). Tensors are L2-resident on MI455X (192MB L2).
// ---------------------------------------------------------------------------
__global__ __launch_bounds__(256) void bn_stats(const float* __restrict__ y,
                                                float* __restrict__ stats,
                                                int N, int C) {
  const int c = blockIdx.x;
  float s = 0.0f, q = 0.0f;
  for (int i = threadIdx.x; i < N; i += 256) {
    const float v = y[(long)i * C + c];
    s += v;
    q += v * v;
  }
  __shared__ float ls[256], lq[256];
  ls[threadIdx.x] = s;
  lq[threadIdx.x] = q;
  __syncthreads();
  for (int off = 128; off > 0; off >>= 1) {
    if (threadIdx.x < off) {
      ls[threadIdx.x] += ls[threadIdx.x + off];
      lq[threadIdx.x] += lq[threadIdx.x + off];
    }
    __syncthreads();
  }
  if (threadIdx.x == 0) {
    const float mu = ls[0] / (float)N;
    const float var = lq[0] / (float)N - mu * mu;
    stats[c] = mu;
    stats[C + c] = rsqrtf(var + 1e-5f);
  }
}

// Fused normalize + affine + optional residual + optional ReLU.
__global__ __launch_bounds__(256) void bn_apply(
    const float* __restrict__ y, const float* __restrict__ stats,
    const float* __restrict__ g, const float* __restrict__ b,
    const float* __restrict__ res, float* __restrict__ out,
    int N, int C, int relu) {
  const long i = (long)blockIdx.x * 256 + threadIdx.x;
  if (i >= (long)N * C) return;
  const int c = (int)(i % C);
  float v = (y[i] - stats[c]) * stats[C + c] * g[c] + b[c];
  if (res) v += res[i];
  if (relu) v = fmaxf(v, 0.0f);
  out[i] = v;
}

__global__ __launch_bounds__(256) void concat2(const float* __restrict__ a, int Ca,
                                               const float* __restrict__ b, int Cb,
                                               float* __restrict__ out, int N) {
  const int C = Ca + Cb;
  const long i = (long)blockIdx.x * 256 + threadIdx.x;
  if (i >= (long)N * C) return;
  const int n = (int)(i / C), c = (int)(i % C);
  out[i] = (c < Ca) ? a[(long)n * Ca + c] : b[(long)n * Cb + (c - Ca)];
}

__global__ __launch_bounds__(256) void final_mm(const float* __restrict__ x,
                                                const float* __restrict__ w,
                                                const float* __restrict__ bias,
                                                float* __restrict__ out,
                                                int N, int Ci, int Co) {
  const int n = blockIdx.x * 256 + threadIdx.x;
  if (n >= N) return;
  for (int j = 0; j < Co; ++j) {
    float s = bias[j];
    for (int i = 0; i < Ci; ++i) s += x[(long)n * Ci + i] * w[i * Co + j];
    out[(long)n * Co + j] = s;
  }
}

// ---------------------------------------------------------------------------
// Host orchestration. d_in is setup_inputs() flattened in dict insertion
// order: feats(0); params c0[1-3] c1[4-6] b1[7-12] c2[13-15] b2[16-24]
// c3[25-27] b3[28-36] c4[37-39] b4[40-48] t4[49-51] b5[52-60] t5[61-63]
// b6[64-72] t6[73-75] b7[76-84] t7[85-87] b8[88-96] final[97-98];
// maps (nbr,mask pairs) m0[99] mb0[101] m1[103] mb1[105] m2[107] mb2[109]
// m3[111] mb3[113] m4[115] mb4[117] mt4[119] mt5[121] mt6[123] mt7[125].
// ---------------------------------------------------------------------------
extern "C" void kernel_launch(void* const* d_in, const int* in_sizes, int n_in,
                              void* d_out, int out_size, void* d_ws, size_t ws_size,
                              hipStream_t stream) {
  auto f = [&](int i) { return (const float*)d_in[i]; };
  auto ip = [&](int i) { return (const int*)d_in[i]; };

  const int N0 = 80000, N1 = 40000, N2 = 20000, N3 = 10000, N4 = 5000;

  // Workspace arena (floats): ~27 MB total.
  float* ws = (float*)d_ws;
  float* STATS = ws;                 // 256
  float* P1 = ws + 256;              // 640000  (out_p1, L0 x 8ch)
  float* B1 = P1 + 640000;           // 320000  (b1, L1 x 8ch)
  float* B2 = B1 + 320000;           // 320000  (b2, L2 x 16ch)
  float* B3 = B2 + 320000;           // 320000  (b3, L3 x 32ch)
  float* XIN = B3 + 320000;          // 1280000 (block input / cat)
  float* TA = XIN + 1280000;         // 1280000
  float* TB = TA + 1280000;          // 1280000
  float* TC = TB + 1280000;          // 1280000

  auto conv = [&](const float* x, const int* nbr, const float* msk, const float* W,
                  float* y, int Nout, int K, int Ci, int Co) {
    const int rowTiles = (Nout + 15) / 16;
    const dim3 g((rowTiles + 3) / 4), blk(128);
    const int cot = (Co + 15) / 16;
    if ((Ci & 3) == 0) {
      if (cot == 1)
        sconv16<1, true><<<g, blk, 0, stream>>>(x, nbr, msk, W, y, Nout, K, Ci, Co);
      else if (cot == 2)
        sconv16<2, true><<<g, blk, 0, stream>>>(x, nbr, msk, W, y, Nout, K, Ci, Co);
      else
        sconv16<4, true><<<g, blk, 0, stream>>>(x, nbr, msk, W, y, Nout, K, Ci, Co);
    } else {
      if (cot == 1)
        sconv16<1, false><<<g, blk, 0, stream>>>(x, nbr, msk, W, y, Nout, K, Ci, Co);
      else if (cot == 2)
        sconv16<2, false><<<g, blk, 0, stream>>>(x, nbr, msk, W, y, Nout, K, Ci, Co);
      else
        sconv16<4, false><<<g, blk, 0, stream>>>(x, nbr, msk, W, y, Nout, K, Ci, Co);
    }
  };
  auto stats = [&](const float* y, int N, int C) {
    bn_stats<<<C, 256, 0, stream>>>(y, STATS, N, C);
  };
  auto bnap = [&](const float* y, const float* g, const float* b, const float* res,
                  float* out, int N, int C, int relu) {
    const long total = (long)N * C;
    bn_apply<<<(int)((total + 255) / 256), 256, 0, stream>>>(y, STATS, g, b, res,
                                                             out, N, C, relu);
  };
  auto cbr = [&](const float* x, int wi, const int* nbr, const float* msk,
                 float* dest, int Nout, int K, int Ci, int Co) {
    conv(x, nbr, msk, f(wi), TA, Nout, K, Ci, Co);
    stats(TA, Nout, Co);
    bnap(TA, f(wi + 1), f(wi + 2), nullptr, dest, Nout, Co, 1);
  };
  auto cat = [&](const float* a, int Ca, const float* b, int Cb, float* out, int N) {
    const long total = (long)N * (Ca + Cb);
    concat2<<<(int)((total + 255) / 256), 256, 0, stream>>>(a, Ca, b, Cb, out, N);
  };
  // BasicBlock: pi = index of w1 (layout w1,g1,b1,w2,g2,b2[,wd,gd,bd]).
  auto block = [&](const float* x, int pi, bool has_wd, const int* nbr,
                   const float* msk, float* dest, int N, int Ci, int Co) {
    conv(x, nbr, msk, f(pi), TA, N, 81, Ci, Co);             // conv1
    stats(TA, N, Co);
    bnap(TA, f(pi + 1), f(pi + 2), nullptr, TB, N, Co, 1);   // u = relu(bn)
    const float* res;
    if (has_wd) {
      conv(x, nullptr, nullptr, f(pi + 6), TA, N, 1, Ci, Co);  // x @ wd
      stats(TA, N, Co);
      bnap(TA, f(pi + 7), f(pi + 8), nullptr, TC, N, Co, 0);   // r = bn
      res = TC;
    } else {
      res = x;
    }
    conv(TB, nbr, msk, f(pi + 3), TA, N, 81, Co, Co);        // conv2(u)
    stats(TA, N, Co);
    bnap(TA, f(pi + 4), f(pi + 5), res, dest, N, Co, 1);     // relu(bn + res)
  };

  // ---- forward graph ----
  cbr(f(0), 1, ip(99), f(100), P1, N0, 125, 1, 8);            // out_p1
  cbr(P1, 4, ip(103), f(104), XIN, N1, 8, 8, 8);              // a1
  block(XIN, 7, false, ip(105), f(106), B1, N1, 8, 8);        // b1
  cbr(B1, 13, ip(107), f(108), XIN, N2, 8, 8, 8);             // a2
  block(XIN, 16, true, ip(109), f(110), B2, N2, 8, 16);       // b2
  cbr(B2, 25, ip(111), f(112), XIN, N3, 8, 16, 16);           // a3
  block(XIN, 28, true, ip(113), f(114), B3, N3, 16, 32);      // b3
  cbr(B3, 37, ip(115), f(116), XIN, N4, 8, 32, 32);           // a4
  block(XIN, 40, true, ip(117), f(118), XIN, N4, 32, 64);     // b4 -> XIN
  cbr(XIN, 49, ip(119), f(120), TB, N3, 8, 64, 64);           // t4 up
  cat(TB, 64, B3, 32, XIN, N3);                               // concat 96ch
  block(XIN, 52, true, ip(113), f(114), XIN, N3, 96, 64);     // b5 (mb3)
  cbr(XIN, 61, ip(121), f(122), TB, N2, 8, 64, 32);           // t5 up
  cat(TB, 32, B2, 16, XIN, N2);                               // concat 48ch
  block(XIN, 64, true, ip(109), f(110), XIN, N2, 48, 32);     // b6 (mb2)
  cbr(XIN, 73, ip(123), f(124), TB, N1, 8, 32, 16);           // t6 up
  cat(TB, 16, B1, 8, XIN, N1);                                // concat 24ch
  block(XIN, 76, true, ip(105), f(106), XIN, N1, 24, 16);     // b7 (mb1)
  cbr(XIN, 85, ip(125), f(126), TB, N0, 8, 16, 8);            // t7 up
  cat(TB, 8, P1, 8, XIN, N0);                                 // concat 16ch
  block(XIN, 88, true, ip(101), f(102), XIN, N0, 16, 8);      // b8 (mb0)
  final_mm<<<(N0 + 255) / 256, 256, 0, stream>>>(XIN, f(97), f(98),
                                                 (float*)d_out, N0, 8, 3);
}